// CotLayer_26439818674594
// MI455X (gfx1250) — compile-verified
//
#include <hip/hip_runtime.h>
#include <math.h>

typedef __attribute__((ext_vector_type(2))) float v2f;
typedef __attribute__((ext_vector_type(8))) float v8f;

#define C_DIM 128
#define H_IMG 256
#define W_IMG 256
#define HW    65536

// D = A(16x4) * B(4x16) + C, fp32 in / fp32 accumulate.
// Operand layout (ISA 7.12.2): reg r of A/B holds K = k0 + r + 2*(lane>=16);
// D reg r holds row M = r + 8*(lane>=16), column N = lane&15.
static __device__ __forceinline__ v8f wmma_f32_16x16x4(v2f a, v2f b, v8f c) {
  return __builtin_amdgcn_wmma_f32_16x16x4_f32(false, a, false, b, (short)0, c,
                                               false, false);
}

// ---------------------------------------------------------------------------
// 1) kemb = relu(grouped 3x3 conv(x, Wk)), groups=4. Implicit-GEMM:
//    per group: [32 x 288] * [288 x Npix]. K ordered as k' = t*32 + ci so a
//    K-quad never straddles a tap -> one (dy,dx) per quad. Zero-pad is done
//    branchlessly: clamp address into bounds, multiply by 0/1 mask (keeps
//    EXEC fully populated, no s_and_saveexec ladders in the WMMA loop).
// grid = (1024, 4 groups, 2 batch), block = 128 (4 waves, 4 pixel tiles)
// ---------------------------------------------------------------------------
__global__ void k_keyembed(const float* __restrict__ x,
                           const float* __restrict__ Wk,
                           float* __restrict__ kemb) {
  extern __shared__ float lds[];                      // 32*288 floats (36 KB)
  const int tid = threadIdx.x;
  const int lane = tid & 31, wave = tid >> 5;
  const int g = blockIdx.y, bb = blockIdx.z;

  const float* Wg = Wk + g * 32 * 288;                // [32 out][32 in][3][3]
  for (int idx = tid; idx < 32 * 288; idx += 128) {   // reorder to [m][t*32+ci]
    int m = idx / 288, kp = idx - m * 288;
    int t = kp >> 5, ci = kp & 31;
    lds[idx] = Wg[m * 288 + ci * 9 + t];
  }
  __syncthreads();

  const int tile = blockIdx.x * 4 + wave;             // 4096 tiles per (g,b)
  const int h = tile >> 4, w0 = (tile & 15) << 4;
  const int hi = lane >> 4, mn = lane & 15;
  const float* xg = x + (bb * C_DIM + g * 32) * HW;

  v8f acc0 = {}, acc1 = {};
  #pragma unroll
  for (int t = 0; t < 9; ++t) {
    const int dy = t / 3 - 1, dx = t % 3 - 1;
    const int hs = h + dy;
    const int wsx = w0 + mn + dx;
    const bool ok = ((unsigned)hs < (unsigned)H_IMG) &&
                    ((unsigned)wsx < (unsigned)W_IMG);
    const float okf = ok ? 1.0f : 0.0f;
    const int hs_c = min(max(hs, 0), H_IMG - 1);      // clamped: always legal
    const int ws_c = min(max(wsx, 0), W_IMG - 1);
    const int rowoff = hs_c * W_IMG + ws_c;
    for (int cb = 0; cb < 32; cb += 4) {
      const int ci = cb + 2 * hi;
      v2f bv;
      bv.x = okf * xg[ci * HW + rowoff];
      bv.y = okf * xg[(ci + 1) * HW + rowoff];
      const int kq = t * 32 + cb + 2 * hi;
      v2f a0, a1;
      a0.x = lds[mn * 288 + kq];        a0.y = lds[mn * 288 + kq + 1];
      a1.x = lds[(16 + mn) * 288 + kq]; a1.y = lds[(16 + mn) * 288 + kq + 1];
      acc0 = wmma_f32_16x16x4(a0, bv, acc0);
      acc1 = wmma_f32_16x16x4(a1, bv, acc1);
    }
  }
  float* outp = kemb + (bb * C_DIM + g * 32) * HW + h * W_IMG + w0 + mn;
  #pragma unroll
  for (int r = 0; r < 8; ++r) {
    int row = r + 8 * hi;
    outp[row * HW]        = fmaxf(acc0[r], 0.0f);
    outp[(16 + row) * HW] = fmaxf(acc1[r], 0.0f);
  }
}

// ---------------------------------------------------------------------------
// 2) Fused embed: t = relu(We1[64x256] @ [x;kemb]); w = We2[144x64] @ t + be2
//    t is routed through per-wave LDS to convert D-layout -> B-layout.
// grid = 1024, block = 256 (8 waves, one 16-pixel tile each)
// ---------------------------------------------------------------------------
__global__ void k_embed(const float* __restrict__ x,
                        const float* __restrict__ kemb,
                        const float* __restrict__ We1,
                        const float* __restrict__ We2,
                        const float* __restrict__ be2,
                        float* __restrict__ wdyn) {
  extern __shared__ float smem[];                     // 33792 floats (132 KB)
  float* We1s = smem;                                 // 64*256
  float* We2s = smem + 64 * 256;                      // 144*64
  float* tb = smem + 64 * 256 + 144 * 64 + (threadIdx.x >> 5) * (64 * 16);
  const int tid = threadIdx.x, lane = tid & 31, wave = tid >> 5;
  for (int idx = tid; idx < 64 * 256; idx += 256) We1s[idx] = We1[idx];
  for (int idx = tid; idx < 144 * 64; idx += 256) We2s[idx] = We2[idx];
  __syncthreads();

  const int tg = blockIdx.x * 8 + wave;               // 0..8191
  const int bb = tg >> 12, tile = tg & 4095;
  const int h = tile >> 4, w0 = (tile & 15) << 4;
  const int hi = lane >> 4, mn = lane & 15;
  const int pix = h * W_IMG + w0 + mn;

  v8f acc[4] = {};
  for (int k0 = 0; k0 < 256; k0 += 4) {               // K over [x;kemb]
    const float* src = (k0 < 128) ? x : kemb;
    const int ch = (k0 & 127) + 2 * hi;
    const int base = (bb * C_DIM + ch) * HW + pix;
    v2f bv; bv.x = src[base]; bv.y = src[base + HW];
    #pragma unroll
    for (int mt = 0; mt < 4; ++mt) {
      v2f av;
      av.x = We1s[(mt * 16 + mn) * 256 + k0 + 2 * hi];
      av.y = We1s[(mt * 16 + mn) * 256 + k0 + 1 + 2 * hi];
      acc[mt] = wmma_f32_16x16x4(av, bv, acc[mt]);
    }
  }
  #pragma unroll
  for (int mt = 0; mt < 4; ++mt)
    #pragma unroll
    for (int r = 0; r < 8; ++r) {
      int row = mt * 16 + r + 8 * hi;
      tb[row * 16 + mn] = fmaxf(acc[mt][r], 0.0f);    // relu, D -> B layout
    }
  __syncthreads();

  v8f acc2[9] = {};
  for (int k0 = 0; k0 < 64; k0 += 4) {
    v2f bv;
    bv.x = tb[(k0 + 2 * hi) * 16 + mn];
    bv.y = tb[(k0 + 1 + 2 * hi) * 16 + mn];
    #pragma unroll
    for (int mt = 0; mt < 9; ++mt) {
      v2f av;
      av.x = We2s[(mt * 16 + mn) * 64 + k0 + 2 * hi];
      av.y = We2s[(mt * 16 + mn) * 64 + k0 + 1 + 2 * hi];
      acc2[mt] = wmma_f32_16x16x4(av, bv, acc2[mt]);
    }
  }
  float* outp = wdyn + (bb * 144) * HW + pix;
  #pragma unroll
  for (int mt = 0; mt < 9; ++mt)
    #pragma unroll
    for (int r = 0; r < 8; ++r) {
      int row = mt * 16 + r + 8 * hi;                 // < 144
      outp[row * HW] = acc2[mt][r] + be2[row];
    }
}

// ---------------------------------------------------------------------------
// 3) xv = Wv[128x128] @ x (1x1 conv).  grid = 1024, block = 256.
// ---------------------------------------------------------------------------
__global__ void k_value(const float* __restrict__ x,
                        const float* __restrict__ Wv,
                        float* __restrict__ xv) {
  extern __shared__ float Wvs[];                      // 128*128 floats (64 KB)
  const int tid = threadIdx.x, lane = tid & 31, wave = tid >> 5;
  for (int idx = tid; idx < 128 * 128; idx += 256) Wvs[idx] = Wv[idx];
  __syncthreads();

  const int tg = blockIdx.x * 8 + wave;
  const int bb = tg >> 12, tile = tg & 4095;
  const int h = tile >> 4, w0 = (tile & 15) << 4;
  const int hi = lane >> 4, mn = lane & 15;
  const int pix = h * W_IMG + w0 + mn;
  const float* xb = x + bb * C_DIM * HW + pix;

  v8f acc[8] = {};
  for (int k0 = 0; k0 < 128; k0 += 4) {
    const int ci = k0 + 2 * hi;
    v2f bv; bv.x = xb[ci * HW]; bv.y = xb[(ci + 1) * HW];
    #pragma unroll
    for (int mt = 0; mt < 8; ++mt) {
      v2f av;
      av.x = Wvs[(mt * 16 + mn) * 128 + k0 + 2 * hi];
      av.y = Wvs[(mt * 16 + mn) * 128 + k0 + 1 + 2 * hi];
      acc[mt] = wmma_f32_16x16x4(av, bv, acc[mt]);
    }
  }
  float* outp = xv + bb * C_DIM * HW + pix;
  #pragma unroll
  for (int mt = 0; mt < 8; ++mt)
    #pragma unroll
    for (int r = 0; r < 8; ++r)
      outp[(mt * 16 + r + 8 * hi) * HW] = acc[mt][r];
}

// ---------------------------------------------------------------------------
// 4) agg = relu(local 3x3 weighted aggregation of xv by dynamic weights).
//    Vectorized x4: each thread computes 4 consecutive pixels. Per 3x3 row:
//    one aligned float4 + two edge scalars (6 values reused across 3 dx
//    shifts x 4 pixels); dynamic weights arrive as aligned float4 (b128).
// grid = 32768, block = 256  (33.5M elements / 4)
// ---------------------------------------------------------------------------
__global__ void k_agg(const float* __restrict__ xv,
                      const float* __restrict__ wdyn,
                      float* __restrict__ agg) {
  const int idx4 = blockIdx.x * 256 + threadIdx.x;    // < 8,388,608
  const int e = idx4 << 2;                            // element base
  const int bc = e >> 16;                             // bb*128 + c
  const int pix0 = e & 65535;                         // multiple of 4
  const int bb = bc >> 7, c = bc & 127;
  const int h = pix0 >> 8, w0v = pix0 & 255;          // w0v % 4 == 0
  const int g = c >> 3;                               // share_planes = 8
  const float* xc = xv + bc * HW;
  const float* wp = wdyn + (bb * 144 + g * 9) * HW + pix0;

  float acc[4] = {0.f, 0.f, 0.f, 0.f};
  #pragma unroll
  for (int r3 = 0; r3 < 3; ++r3) {                    // dy = r3-1
    const int hs = h + r3 - 1;
    const float rowm = ((unsigned)hs < 256u) ? 1.0f : 0.0f;
    const int hs_c = min(max(hs, 0), 255);
    const float* xr = xc + hs_c * 256;
    float v[6];
    const float4 xm = *(const float4*)(xr + w0v);
    v[0] = (w0v > 0) ? (rowm * xr[w0v - 1]) : 0.0f;   // left halo (pad 0)
    v[1] = rowm * xm.x; v[2] = rowm * xm.y;
    v[3] = rowm * xm.z; v[4] = rowm * xm.w;
    v[5] = (w0v < 252) ? (rowm * xr[w0v + 4]) : 0.0f; // right halo (pad 0)
    #pragma unroll
    for (int dx = 0; dx < 3; ++dx) {
      const int t = r3 * 3 + dx;
      const float4 wv = *(const float4*)(wp + t * HW);
      acc[0] += v[dx + 0] * wv.x;
      acc[1] += v[dx + 1] * wv.y;
      acc[2] += v[dx + 2] * wv.z;
      acc[3] += v[dx + 3] * wv.w;
    }
  }
  float4 o;
  o.x = fmaxf(acc[0], 0.f); o.y = fmaxf(acc[1], 0.f);
  o.z = fmaxf(acc[2], 0.f); o.w = fmaxf(acc[3], 0.f);
  *(float4*)(agg + e) = o;
}

// ---------------------------------------------------------------------------
// 5) gap[b,c] = mean_{h,w}(agg + kemb).  One block per (b,c), float4 loads.
// ---------------------------------------------------------------------------
__global__ void k_gap(const float* __restrict__ agg,
                      const float* __restrict__ kemb,
                      float* __restrict__ gap) {
  __shared__ float red[256];
  const int bc = blockIdx.x;
  const int base = bc << 16;
  float s = 0.0f;
  for (int i = threadIdx.x << 2; i < HW; i += 1024) {
    const float4 a = *(const float4*)(agg + base + i);
    const float4 k = *(const float4*)(kemb + base + i);
    s += (a.x + k.x) + (a.y + k.y) + (a.z + k.z) + (a.w + k.w);
  }
  red[threadIdx.x] = s;
  __syncthreads();
  for (int off = 128; off > 0; off >>= 1) {
    if (threadIdx.x < off) red[threadIdx.x] += red[threadIdx.x + off];
    __syncthreads();
  }
  if (threadIdx.x == 0) gap[bc] = red[0] * (1.0f / 65536.0f);
}

// ---------------------------------------------------------------------------
// 6) SE MLP + pairwise softmax -> attn[b,c,2].  Single block.
// ---------------------------------------------------------------------------
__global__ void k_se(const float* __restrict__ gap,
                     const float* __restrict__ Ws1, const float* __restrict__ bs1,
                     const float* __restrict__ Ws2, const float* __restrict__ bs2,
                     float* __restrict__ attn) {
  __shared__ float a1s[64];
  __shared__ float a2s[256];
  const int tid = threadIdx.x;
  for (int bb = 0; bb < 2; ++bb) {
    if (tid < 64) {
      float s = bs1[tid];
      for (int c = 0; c < 128; ++c) s += Ws1[tid * 128 + c] * gap[bb * 128 + c];
      a1s[tid] = fmaxf(s, 0.0f);
    }
    __syncthreads();
    {
      float s = bs2[tid];
      for (int j = 0; j < 64; ++j) s += Ws2[tid * 64 + j] * a1s[j];
      a2s[tid] = s;
    }
    __syncthreads();
    if (tid < 128) {                                  // reshape (C,2): pairs
      float v0 = a2s[2 * tid], v1 = a2s[2 * tid + 1];
      float mx = fmaxf(v0, v1);
      float e0 = __expf(v0 - mx), e1 = __expf(v1 - mx);
      float inv = 1.0f / (e0 + e1);
      attn[(bb * 128 + tid) * 2]     = e0 * inv;
      attn[(bb * 128 + tid) * 2 + 1] = e1 * inv;
    }
    __syncthreads();
  }
}

// ---------------------------------------------------------------------------
// 7) out = agg*attn0 + kemb*attn1  (float4 vectorized)
// ---------------------------------------------------------------------------
__global__ void k_out(const float* __restrict__ agg,
                      const float* __restrict__ kemb,
                      const float* __restrict__ attn,
                      float* __restrict__ out) {
  const int idx4 = blockIdx.x * 256 + threadIdx.x;
  const int e = idx4 << 2;
  const int bc = e >> 16;
  const float a0 = attn[bc * 2], a1 = attn[bc * 2 + 1];
  const float4 a = *(const float4*)(agg + e);
  const float4 k = *(const float4*)(kemb + e);
  float4 o;
  o.x = a.x * a0 + k.x * a1;
  o.y = a.y * a0 + k.y * a1;
  o.z = a.z * a0 + k.z * a1;
  o.w = a.w * a0 + k.w * a1;
  *(float4*)(out + e) = o;
}

extern "C" void kernel_launch(void* const* d_in, const int* in_sizes, int n_in,
                              void* d_out, int out_size, void* d_ws,
                              size_t ws_size, hipStream_t stream) {
  (void)in_sizes; (void)n_in; (void)out_size; (void)ws_size;
  const float* x   = (const float*)d_in[0];
  const float* Wk  = (const float*)d_in[1];
  const float* We1 = (const float*)d_in[2];
  const float* We2 = (const float*)d_in[3];
  const float* be2 = (const float*)d_in[4];
  const float* Wv  = (const float*)d_in[5];
  const float* Ws1 = (const float*)d_in[6];
  const float* bs1 = (const float*)d_in[7];
  const float* Ws2 = (const float*)d_in[8];
  const float* bs2 = (const float*)d_in[9];
  float* out = (float*)d_out;

  const size_t NIMG = (size_t)2 * C_DIM * HW;         // 16,777,216 floats
  float* kemb = (float*)d_ws;
  float* wdyn = kemb + NIMG;                          // 2*144*65536 floats
  float* xv   = wdyn + (size_t)2 * 144 * HW;
  float* agg  = xv + NIMG;
  float* gap  = agg + NIMG;                           // 256 floats
  float* attn = gap + 256;                            // 512 floats

  k_keyembed<<<dim3(1024, 4, 2), 128, 32 * 288 * sizeof(float), stream>>>(
      x, Wk, kemb);
  k_embed<<<dim3(1024), 256,
            (64 * 256 + 144 * 64 + 8 * 64 * 16) * sizeof(float), stream>>>(
      x, kemb, We1, We2, be2, wdyn);
  k_value<<<dim3(1024), 256, 128 * 128 * sizeof(float), stream>>>(x, Wv, xv);
  k_agg<<<dim3(32768), 256, 0, stream>>>(xv, wdyn, agg);
  k_gap<<<dim3(256), 256, 0, stream>>>(agg, kemb, gap);
  k_se<<<dim3(1), 256, 0, stream>>>(gap, Ws1, bs1, Ws2, bs2, attn);
  k_out<<<dim3(32768), 256, 0, stream>>>(agg, kemb, attn, out);
}